// GnnNodeClassifier_83588653515065
// MI455X (gfx1250) — compile-verified
//
#include <hip/hip_runtime.h>
#include <hip/hip_bf16.h>
#include <math.h>

typedef float v2f __attribute__((ext_vector_type(2)));
typedef float v8f __attribute__((ext_vector_type(8)));

#define CHID 64          // hidden channels per head
#define NEG_SLOPE 0.2f
#define EPSV 1e-16f

// ---------------------------------------------------------------- utilities
__global__ void fill_f32(float* __restrict__ p, float v, int n) {
    int i = blockIdx.x * blockDim.x + threadIdx.x;
    if (i < n) p[i] = v;
}

__device__ __forceinline__ void atomicMaxF(float* a, float v) {
    // classic monotonic-bit-pattern trick; memory pre-initialized to -inf
    if (v >= 0.0f) atomicMax((int*)a, __float_as_int(v));
    else           atomicMin((unsigned int*)a, __float_as_uint(v));
}

__device__ __forceinline__ void edge_nodes(const int* __restrict__ ei, int E0,
                                           int e, int& s, int& d) {
    if (e < E0) { s = ei[e]; d = ei[E0 + e]; }
    else        { s = e - E0; d = s; }      // self loops appended
}

// ---------------------------------------------------------------- WMMA GEMM
// Hout[M,NC] = X[M,K] @ W[K,NC], f32, one wave per 16x64 output tile.
// grid = (M/16, NC/64), block = 32 (one wave; EXEC all-ones, no divergence).
// A fragment (16x4 f32): lane L: m=L&15, regs = A[m][2*(L>>4) + {0,1}]
// B fragment (4x16 f32): lane L: n=L&15, regs = B[2*(L>>4) + {0,1}][n]
// C/D (16x16 f32): lane L: n=L&15, reg v holds D[8*(L>>4)+v][n]
__global__ void gemm_wmma(const float* __restrict__ X, const float* __restrict__ W,
                          float* __restrict__ Hout, int K, int NC) {
    int lane = threadIdx.x;
    int half = lane >> 4;
    int l16  = lane & 15;
    int mbase = blockIdx.x * 16;
    int nbase = blockIdx.y * 64 + l16;
    const float* xrow = X + (size_t)(mbase + l16) * K;

    v8f acc0 = {}, acc1 = {}, acc2 = {}, acc3 = {};
    for (int k = 0; k < K; k += 4) {
        int ka = k + half * 2;
        v2f a; a.x = xrow[ka]; a.y = xrow[ka + 1];
        const float* w0 = W + (size_t)ka * NC + nbase;
        const float* w1 = w0 + NC;
        v2f b0; b0.x = w0[0];  b0.y = w1[0];
        v2f b1; b1.x = w0[16]; b1.y = w1[16];
        v2f b2; b2.x = w0[32]; b2.y = w1[32];
        v2f b3; b3.x = w0[48]; b3.y = w1[48];
        acc0 = __builtin_amdgcn_wmma_f32_16x16x4_f32(false, a, false, b0, (short)0, acc0, false, false);
        acc1 = __builtin_amdgcn_wmma_f32_16x16x4_f32(false, a, false, b1, (short)0, acc1, false, false);
        acc2 = __builtin_amdgcn_wmma_f32_16x16x4_f32(false, a, false, b2, (short)0, acc2, false, false);
        acc3 = __builtin_amdgcn_wmma_f32_16x16x4_f32(false, a, false, b3, (short)0, acc3, false, false);
    }
    for (int v = 0; v < 8; ++v) {
        int row = mbase + half * 8 + v;
        float* hr = Hout + (size_t)row * NC + nbase;
        hr[0]  = acc0[v];
        hr[16] = acc1[v];
        hr[32] = acc2[v];
        hr[48] = acc3[v];
    }
}

// ------------------------------------------------- per-node attention logits
// as_[n,h] = <h[n,h,:], a_src[h,:]>, ad_ likewise. One thread per (n,h).
__global__ void att_logits(const float* __restrict__ Hf,
                           const float* __restrict__ a_src,
                           const float* __restrict__ a_dst,
                           float* __restrict__ as_, float* __restrict__ ad_,
                           int N, int NH) {
    int t = blockIdx.x * blockDim.x + threadIdx.x;
    if (t >= N * NH) return;
    int n = t / NH, h = t % NH;
    const float* hp = Hf + (size_t)n * NH * CHID + (size_t)h * CHID;
    float s = 0.f, d = 0.f;
    for (int c = 0; c < CHID; ++c) {
        float hv = hp[c];
        s += hv * a_src[h * CHID + c];
        d += hv * a_dst[h * CHID + c];
    }
    as_[t] = s; ad_[t] = d;
}

// --------------------------------------------------------------- edge passes
__global__ void edge_max(const int* __restrict__ ei, int E0, int Et,
                         const float* __restrict__ as_, const float* __restrict__ ad_,
                         float* __restrict__ m, int NH) {
    int e = blockIdx.x * blockDim.x + threadIdx.x;
    if (e >= Et) return;
    int s, d; edge_nodes(ei, E0, e, s, d);
    for (int h = 0; h < NH; ++h) {
        float x = as_[s * NH + h] + ad_[d * NH + h];
        x = (x >= 0.f) ? x : NEG_SLOPE * x;
        atomicMaxF(&m[d * NH + h], x);
    }
}

__global__ void edge_expsum(const int* __restrict__ ei, int E0, int Et,
                            const float* __restrict__ as_, const float* __restrict__ ad_,
                            const float* __restrict__ m, float* __restrict__ den, int NH) {
    int e = blockIdx.x * blockDim.x + threadIdx.x;
    if (e >= Et) return;
    int s, d; edge_nodes(ei, E0, e, s, d);
    for (int h = 0; h < NH; ++h) {
        float x = as_[s * NH + h] + ad_[d * NH + h];
        x = (x >= 0.f) ? x : NEG_SLOPE * x;
        atomicAdd(&den[d * NH + h], expf(x - m[d * NH + h]));
    }
}

__global__ void edge_alpha(const int* __restrict__ ei, int E0, int Et,
                           const float* __restrict__ as_, const float* __restrict__ ad_,
                           const float* __restrict__ m, const float* __restrict__ den,
                           float* __restrict__ alpha, int NH) {
    int t = blockIdx.x * blockDim.x + threadIdx.x;
    if (t >= Et * NH) return;
    int e = t / NH, h = t % NH;
    int s, d; edge_nodes(ei, E0, e, s, d);
    float x = as_[s * NH + h] + ad_[d * NH + h];
    x = (x >= 0.f) ? x : NEG_SLOPE * x;
    float ex = expf(x - m[d * NH + h]);
    alpha[t] = ex / (den[d * NH + h] + EPSV);
}

// message scatter: one thread per (edge, head*channel); L2-resident atomics
__global__ void edge_scatter(const int* __restrict__ ei, int E0, int Et,
                             const float* __restrict__ Hf, const float* __restrict__ alpha,
                             float* __restrict__ acc, int NH) {
    int HC = NH * CHID;
    int t = blockIdx.x * blockDim.x + threadIdx.x;   // <= 115.2M, fits int
    if (t >= Et * HC) return;
    int e = t / HC, r = t % HC, h = r / CHID;
    int s, d; edge_nodes(ei, E0, e, s, d);
    float msg = Hf[(size_t)s * HC + r] * alpha[e * NH + h];
    atomicAdd(&acc[(size_t)d * HC + r], msg);
}

// ------------------------------------------------------------- pointwise ops
__global__ void bias_elu(float* __restrict__ acc, const float* __restrict__ b,
                         int N, int F) {
    int t = blockIdx.x * blockDim.x + threadIdx.x;
    if (t >= N * F) return;
    float v = acc[t] + b[t % F];
    acc[t] = (v > 0.f) ? v : (expf(v) - 1.f);
}

// out[n,:] = elu(acc2[n,:] + b2) @ Wout + bout   (Wout: [64,3] row-major)
__global__ void final_out(const float* __restrict__ acc2, const float* __restrict__ b2,
                          const float* __restrict__ Wout, const float* __restrict__ bout,
                          float* __restrict__ out, int N) {
    int n = blockIdx.x * blockDim.x + threadIdx.x;
    if (n >= N) return;
    float o0 = bout[0], o1 = bout[1], o2 = bout[2];
    for (int c = 0; c < CHID; ++c) {
        float v = acc2[(size_t)n * CHID + c] + b2[c];
        v = (v > 0.f) ? v : (expf(v) - 1.f);
        o0 += v * Wout[c * 3 + 0];
        o1 += v * Wout[c * 3 + 1];
        o2 += v * Wout[c * 3 + 2];
    }
    out[n * 3 + 0] = o0; out[n * 3 + 1] = o1; out[n * 3 + 2] = o2;
}

// ------------------------------------------------------------------- driver
extern "C" void kernel_launch(void* const* d_in, const int* in_sizes, int n_in,
                              void* d_out, int out_size, void* d_ws, size_t ws_size,
                              hipStream_t stream) {
    const float* x    = (const float*)d_in[0];
    const int*   ei   = (const int*)  d_in[1];
    const float* W1   = (const float*)d_in[2];
    const float* as1  = (const float*)d_in[3];
    const float* ad1  = (const float*)d_in[4];
    const float* b1   = (const float*)d_in[5];
    const float* W2   = (const float*)d_in[6];
    const float* as2  = (const float*)d_in[7];
    const float* ad2  = (const float*)d_in[8];
    const float* b2   = (const float*)d_in[9];
    const float* Wout = (const float*)d_in[10];
    const float* bout = (const float*)d_in[11];
    float* out = (float*)d_out;

    const int N  = in_sizes[0] / 128;   // 50000 (divisible by 16)
    const int E0 = in_sizes[1] / 2;     // 400000
    const int Et = E0 + N;              // + self loops

    // workspace layout (floats) -- everything (~113MB) fits in the 192MB L2
    float* ws     = (float*)d_ws;
    float* hbuf   = ws;                          // N*256 : h1 (later h2 / acc2)
    float* accb   = hbuf + (size_t)N * 256;      // N*256 : gat1 out -> x2
    float* asb    = accb + (size_t)N * 256;      // N*4
    float* adb    = asb  + (size_t)N * 4;        // N*4
    float* mb     = adb  + (size_t)N * 4;        // N*4
    float* denb   = mb   + (size_t)N * 4;        // N*4
    float* alphab = denb + (size_t)N * 4;        // Et*4

    const int TB = 256;
    auto nb = [](long long n, int tb) { return (unsigned)((n + tb - 1) / tb); };

    // ---------------- layer 1: GATConv(128 -> 64, heads=4) ----------------
    gemm_wmma<<<dim3(N / 16, 256 / 64), 32, 0, stream>>>(x, W1, hbuf, 128, 256);
    att_logits<<<nb((long long)N * 4, TB), TB, 0, stream>>>(hbuf, as1, ad1, asb, adb, N, 4);
    fill_f32<<<nb(N * 4, TB), TB, 0, stream>>>(mb,   -INFINITY, N * 4);
    fill_f32<<<nb(N * 4, TB), TB, 0, stream>>>(denb, 0.0f,      N * 4);
    fill_f32<<<nb((long long)N * 256, TB), TB, 0, stream>>>(accb, 0.0f, N * 256);
    edge_max   <<<nb(Et, TB), TB, 0, stream>>>(ei, E0, Et, asb, adb, mb, 4);
    edge_expsum<<<nb(Et, TB), TB, 0, stream>>>(ei, E0, Et, asb, adb, mb, denb, 4);
    edge_alpha <<<nb((long long)Et * 4, TB), TB, 0, stream>>>(ei, E0, Et, asb, adb, mb, denb, alphab, 4);
    edge_scatter<<<nb((long long)Et * 256, TB), TB, 0, stream>>>(ei, E0, Et, hbuf, alphab, accb, 4);
    bias_elu<<<nb((long long)N * 256, TB), TB, 0, stream>>>(accb, b1, N, 256);

    // ---------------- layer 2: GATConv(256 -> 64, heads=1) ----------------
    float* h2   = hbuf;                       // reuse: [N,64]
    float* acc2 = hbuf + (size_t)N * 64;      // reuse: [N,64]
    gemm_wmma<<<dim3(N / 16, 1), 32, 0, stream>>>(accb, W2, h2, 256, 64);
    att_logits<<<nb(N, TB), TB, 0, stream>>>(h2, as2, ad2, asb, adb, N, 1);
    fill_f32<<<nb(N, TB), TB, 0, stream>>>(mb,   -INFINITY, N);
    fill_f32<<<nb(N, TB), TB, 0, stream>>>(denb, 0.0f,      N);
    fill_f32<<<nb((long long)N * 64, TB), TB, 0, stream>>>(acc2, 0.0f, N * 64);
    edge_max   <<<nb(Et, TB), TB, 0, stream>>>(ei, E0, Et, asb, adb, mb, 1);
    edge_expsum<<<nb(Et, TB), TB, 0, stream>>>(ei, E0, Et, asb, adb, mb, denb, 1);
    edge_alpha <<<nb(Et, TB), TB, 0, stream>>>(ei, E0, Et, asb, adb, mb, denb, alphab, 1);
    edge_scatter<<<nb((long long)Et * 64, TB), TB, 0, stream>>>(ei, E0, Et, h2, alphab, acc2, 1);

    // ---------------- head: elu -> @Wout + bout ----------------
    final_out<<<nb(N, TB), TB, 0, stream>>>(acc2, b2, Wout, bout, out, N);
}